// DiceLoss_34239479283999
// MI455X (gfx1250) — compile-verified
//
#include <hip/hip_runtime.h>

#define HW (512 * 1024)      // pixels per image
#define NCLS 19
#define NB 4

typedef float v2f __attribute__((ext_vector_type(2)));
typedef float v8f __attribute__((ext_vector_type(8)));

// ---------------------------------------------------------------------------
// Kernel 1: zero the global histogram region of the workspace each launch
// (graph-capture safe, deterministic). Layout: [0..31]=pred, [32..63]=label,
// [64..95]=intersection.
// ---------------------------------------------------------------------------
__global__ void iou_init(unsigned* __restrict__ hist) {
    int t = threadIdx.x;
    if (t < 96) hist[t] = 0u;
}

// ---------------------------------------------------------------------------
// Kernel 2: streaming argmax + histograms.
// Each thread owns 4 consecutive pixels: 19 coalesced float4 (b128) loads for
// the logits, 1 int4 load for the labels, strict '>' argmax (first-occurrence
// tie-break matches jnp.argmax), LDS-privatized histograms via ds atomics,
// per-block flush with global atomics.
// ---------------------------------------------------------------------------
__global__ __launch_bounds__(256) void iou_main(const float* __restrict__ input,
                                                const int* __restrict__ target,
                                                unsigned* __restrict__ hist) {
    __shared__ unsigned lh[64];   // [0..18]=pred, [20..38]=label, [40..58]=inter
    const int tid = threadIdx.x;
    if (tid < 64) lh[tid] = 0u;
    __syncthreads();

    const size_t gid = (size_t)blockIdx.x * blockDim.x + tid;   // 524288 groups
    const int b = (int)(gid >> 17);              // gid / (HW/4), HW/4 = 2^17
    const int g = (int)(gid & (HW / 4 - 1));

    const float4* p0 =
        reinterpret_cast<const float4*>(input) + (size_t)b * NCLS * (HW / 4) + g;

    float4 best = p0[0];
    int bx = 0, by = 0, bz = 0, bw = 0;
#pragma unroll
    for (int c = 1; c < NCLS; ++c) {
        float4 v = p0[(size_t)c * (HW / 4)];
        if (v.x > best.x) { best.x = v.x; bx = c; }
        if (v.y > best.y) { best.y = v.y; by = c; }
        if (v.z > best.z) { best.z = v.z; bz = c; }
        if (v.w > best.w) { best.w = v.w; bw = c; }
    }

    const int4 tq =
        reinterpret_cast<const int4*>(target)[(size_t)b * (HW / 4) + g];

    atomicAdd(&lh[bx], 1u);
    atomicAdd(&lh[by], 1u);
    atomicAdd(&lh[bz], 1u);
    atomicAdd(&lh[bw], 1u);
    atomicAdd(&lh[20 + tq.x], 1u);
    atomicAdd(&lh[20 + tq.y], 1u);
    atomicAdd(&lh[20 + tq.z], 1u);
    atomicAdd(&lh[20 + tq.w], 1u);
    if (bx == tq.x) atomicAdd(&lh[40 + bx], 1u);
    if (by == tq.y) atomicAdd(&lh[40 + by], 1u);
    if (bz == tq.z) atomicAdd(&lh[40 + bz], 1u);
    if (bw == tq.w) atomicAdd(&lh[40 + bw], 1u);

    __syncthreads();
    if (tid < 60) {
        const int grp = tid / 20;
        const int c = tid % 20;
        const unsigned v = lh[tid];
        if (v) atomicAdd(&hist[grp * 32 + c], v);
    }
}

// ---------------------------------------------------------------------------
// Kernel 3: one wave32. Lane c computes IoU_c, then an exact f32 cross-lane
// sum via V_WMMA_F32_16X16X4_F32:
//   A[m][0] = iou_m, A[m][2] = iou_{m+16} (lane l -> {VGPR0=iou, VGPR1=0}),
//   B = all ones  =>  D[m][n] = iou_m + iou_{m+16}.
// Summing a lane's 8 D VGPRs and combining lane 0 + lane 16 gives sum(IoU).
// Branchless before the WMMA so EXEC is all 1s.
// ---------------------------------------------------------------------------
__global__ void iou_final(const unsigned* __restrict__ hist,
                          float* __restrict__ out) {
    const int lane = threadIdx.x;   // 32 lanes
    const float pred  = (float)hist[lane];
    const float label = (float)hist[32 + lane];
    const float inter = (float)hist[64 + lane];
    const float uni   = pred + label - inter;
    // NaN-free select (lanes >= 19 contribute exactly 0)
    const float num   = (lane < NCLS) ? inter : 0.0f;
    const float den   = (lane < NCLS) ? uni : 1.0f;
    const float iou   = num / den;

#if __has_builtin(__builtin_amdgcn_wmma_f32_16x16x4_f32)
    v2f a;  a[0] = iou;  a[1] = 0.0f;
    v2f bm; bm[0] = 1.0f; bm[1] = 1.0f;          // B matrix = all ones
    v8f c = {0.f, 0.f, 0.f, 0.f, 0.f, 0.f, 0.f, 0.f};
    v8f d = __builtin_amdgcn_wmma_f32_16x16x4_f32(
        /*neg_a=*/false, a, /*neg_b=*/false, bm,
        /*c_mod=*/(short)0, c, /*reuse_a=*/false, /*reuse_b=*/false);
    float t = d[0] + d[1] + d[2] + d[3] + d[4] + d[5] + d[6] + d[7];
    float total = __shfl(t, 0, 32) + __shfl(t, 16, 32);
#else
    float total = iou;
#pragma unroll
    for (int off = 16; off > 0; off >>= 1) total += __shfl_xor(total, off, 32);
#endif

    if (lane == 0) out[0] = total * (1.0f / (float)NCLS);
}

// ---------------------------------------------------------------------------
extern "C" void kernel_launch(void* const* d_in, const int* in_sizes, int n_in,
                              void* d_out, int out_size, void* d_ws, size_t ws_size,
                              hipStream_t stream) {
    const float* input  = (const float*)d_in[0];
    const int*   target = (const int*)d_in[1];
    float*       out    = (float*)d_out;
    unsigned*    hist   = (unsigned*)d_ws;   // needs 96 * 4 = 384 bytes

    hipLaunchKernelGGL(iou_init, dim3(1), dim3(128), 0, stream, hist);

    const int groups = NB * (HW / 4);        // 524288 four-pixel groups
    hipLaunchKernelGGL(iou_main, dim3(groups / 256), dim3(256), 0, stream,
                       input, target, hist);

    hipLaunchKernelGGL(iou_final, dim3(1), dim3(32), 0, stream, hist, out);
}